// TSCoupler_979252544276
// MI455X (gfx1250) — compile-verified
//
#include <hip/hip_runtime.h>

#define B_N 64
#define L_N 8192
#define C_N 21
#define K_N 15
#define TPB 256

typedef float v2f __attribute__((ext_vector_type(2)));
typedef float v8f __attribute__((ext_vector_type(8)));
typedef unsigned int u32x4 __attribute__((ext_vector_type(4)));
typedef int i32x4 __attribute__((ext_vector_type(4)));
typedef int i32x8 __attribute__((ext_vector_type(8)));

#define PI_F 3.14159265358979323846f
#define TWO_PI_F 6.28318530717958647692f

// ---------------------------------------------------------------------------
// V_WMMA_F32_16X16X4_F32 wrapper: D(16x16,f32) = A(16x4) * B(4x16) + C
// ---------------------------------------------------------------------------
static __device__ __forceinline__ v8f wmma_f32_16x16x4(v2f a, v2f b, v8f c) {
#if defined(__HIP_DEVICE_COMPILE__)
  // 8 args: (neg_a, A, neg_b, B, c_mod, C, reuse_a, reuse_b)
  return __builtin_amdgcn_wmma_f32_16x16x4_f32(false, a, false, b, (short)0, c,
                                               false, false);
#else
  (void)a; (void)b;
  return c;
#endif
}

static __device__ __forceinline__ float softplusf(float x) {
  return (x > 20.f) ? x : log1pf(expf(x));
}

// d_mod = mod(d+pi, 2pi) - pi, with jnp edge case (d_mod==-pi && d>0) -> pi
static __device__ __forceinline__ float wrap_dm(float d) {
  float m = fmodf(d + PI_F, TWO_PI_F);
  if (m < 0.f) m += TWO_PI_F;
  float dm = m - PI_F;
  if (dm == -PI_F && d > 0.f) dm = PI_F;
  return dm;
}

// ---------------------------------------------------------------------------
// One radix-16 Stockham DIF stage over the full 8192-point working set.
//   for p in [0,m), q in [0,s):  b_k = sum_r x[q+s*(p+m*r)] * W16^{rk}
//                                y[q+s*(16p+k)] = b_k * Wn^{pk}
// Columns j = q + s*p  (j in [0,512), since m*s = N/16 = 512 at every stage)
//   => B loads are x[j + 512*r]; output base = 16*j - 15*q.
// The 16-point DFT across 16 columns is one f32 WMMA tile (K chained 4x4).
// ---------------------------------------------------------------------------
static __device__ __forceinline__ void radix16_stage(
    const float* xr, const float* xi, float* yr, float* yi, int n, int slog,
    const v2f* fAr, const v2f* fAiN, const v2f* fAi, int lane, int wave) {
  const int s = 1 << slog;
  const int col = lane & 15;
  const int hi = lane >> 4;
  const float tw0 = -TWO_PI_F / (float)n;
  for (int tile = wave; tile < 32; tile += 8) {  // 8 waves x 4 tiles
    const int j = tile * 16 + col;
    v8f cr = {0.f, 0.f, 0.f, 0.f, 0.f, 0.f, 0.f, 0.f};
    v8f ci = {0.f, 0.f, 0.f, 0.f, 0.f, 0.f, 0.f, 0.f};
#pragma unroll
    for (int kc = 0; kc < 4; ++kc) {
      const int r0 = 4 * kc + 2 * hi;  // B rows per ISA K-split layout
      v2f br = {xr[j + 512 * r0], xr[j + 512 * (r0 + 1)]};
      v2f bi = {xi[j + 512 * r0], xi[j + 512 * (r0 + 1)]};
      // Yr = Fr*Xr - Fi*Xi ; Yi = Fr*Xi + Fi*Xr
      cr = wmma_f32_16x16x4(fAr[kc], br, cr);
      cr = wmma_f32_16x16x4(fAiN[kc], bi, cr);
      ci = wmma_f32_16x16x4(fAr[kc], bi, ci);
      ci = wmma_f32_16x16x4(fAi[kc], br, ci);
    }
    const int q = j & (s - 1);
    const int p = j >> slog;
    const int obase = 16 * j - 15 * q;
#pragma unroll
    for (int v = 0; v < 8; ++v) {
      const int k = v + 8 * hi;  // C/D row mapping (ISA 7.12.2)
      float sw, cw;
      __sincosf(tw0 * (float)(p * k), &sw, &cw);
      const float vr = cr[v], vi = ci[v];
      yr[obase + s * k] = vr * cw - vi * sw;
      yi[obase + s * k] = vr * sw + vi * cw;
    }
  }
}

// Forward 8192-pt FFT: radix 16,16,16,2. In: (r0,i0). Result back in (r0,i0).
static __device__ __forceinline__ void fft8192(float* r0, float* i0, float* r1,
                                               float* i1, const v2f* fAr,
                                               const v2f* fAiN, const v2f* fAi,
                                               int lane, int wave, int tid) {
  __syncthreads();
  radix16_stage(r0, i0, r1, i1, 8192, 0, fAr, fAiN, fAi, lane, wave);
  __syncthreads();
  radix16_stage(r1, i1, r0, i0, 512, 4, fAr, fAiN, fAi, lane, wave);
  __syncthreads();
  radix16_stage(r0, i0, r1, i1, 32, 8, fAr, fAiN, fAi, lane, wave);
  __syncthreads();
  for (int t = tid; t < 4096; t += TPB) {  // final radix-2 (n=2, s=4096)
    const float ar = r1[t], ai = i1[t];
    const float br = r1[t + 4096], bi = i1[t + 4096];
    r0[t] = ar + br;        i0[t] = ai + bi;
    r0[t + 4096] = ar - br; i0[t + 4096] = ai - bi;
  }
  __syncthreads();
}

__global__ __launch_bounds__(TPB, 1) void tsc_kernel(
    const float* __restrict__ seasonal, const float* __restrict__ trend,
    const float* __restrict__ pcw, const float* __restrict__ strength,
    const float* __restrict__ alpha_log, const float* __restrict__ phi0,
    const float* __restrict__ b1l, const float* __restrict__ b2l,
    float* __restrict__ out) {
  extern __shared__ float smem[];
  float* r0 = smem;            // 8192
  float* i0 = smem + 8192;     // 8192
  float* r1 = smem + 16384;    // 8192
  float* i1 = smem + 24576;    // 8192
  float* scanb = smem + 32768; // 256
  float* wz = smem + 33024;    // 16

  const int tid = threadIdx.x;
  const int lane = tid & 31;
  const int wave = tid >> 5;
  const int b = blockIdx.x / C_N;
  const int c = blockIdx.x % C_N;

  // --- DFT-16 matrix in WMMA A-fragment layout (per lane, registers) ------
  v2f fAr[4], fAi[4], fAiN[4];
  {
    const int mrow = lane & 15;
    const int hi = lane >> 4;
#pragma unroll
    for (int kc = 0; kc < 4; ++kc) {
#pragma unroll
      for (int jj = 0; jj < 2; ++jj) {
        const int r = 4 * kc + 2 * hi + jj;
        const float th = -(TWO_PI_F / 16.f) * (float)((mrow * r) & 15);
        float s_, c_;
        __sincosf(th, &s_, &c_);
        fAr[kc][jj] = c_;
        fAi[kc][jj] = s_;
        fAiN[kc][jj] = -s_;
      }
    }
  }

  // --- zero imag plane -----------------------------------------------------
  for (int t = tid; t < L_N; t += TPB) i0[t] = 0.f;

  // --- zero-mean depthwise conv weights -----------------------------------
  if (tid == 0) {
    float mean = 0.f;
    for (int jj = 0; jj < K_N; ++jj) mean += pcw[c * K_N + jj];
    mean *= (1.f / (float)K_N);
    for (int jj = 0; jj < K_N; ++jj) wz[jj] = pcw[c * K_N + jj] - mean;
  }

  // --- gather the stride-C channel signal into LDS via the TDM ------------
#if defined(__HIP_DEVICE_COMPILE__) && \
    __has_builtin(__builtin_amdgcn_tensor_load_to_lds)
  if (wave == 0) {
    const float* gsrc = seasonal + (size_t)b * (L_N * C_N) + c;
    const unsigned long long ga = (unsigned long long)(size_t)gsrc;
    const unsigned ldsoff = (unsigned)(size_t)(void*)r0;
    // D# group 0: count=1 | lds_addr | global_addr | type=2
    u32x4 g0;
    g0[0] = 1u;
    g0[1] = ldsoff;
    g0[2] = (unsigned)(ga & 0xffffffffull);
    g0[3] = (unsigned)((ga >> 32) & 0x01ffffffull) | (2u << 30);
    // D# group 1: data_size=4B, tensor 21 x 8192, tile 1 x 8192, stride0=21
    i32x8 g1;
    g1[0] = (int)(2u << 16);             // data_size = 2 (4 bytes)
    g1[1] = (int)((unsigned)C_N << 16);  // tensor_dim0[15:0] @ [31:16]
    g1[2] = (int)((unsigned)L_N << 16);  // dim0 hi=0 ; tensor_dim1 lo @ [31:16]
    g1[3] = (int)(1u << 16);             // dim1 hi=0 ; tile_dim0 = 1
    g1[4] = L_N;                         // tile_dim1 = 8192 ; tile_dim2 = 0
    g1[5] = C_N;                         // tensor_dim0_stride[31:0] = 21
    g1[6] = 0;                           // stride0 hi ; dim1_stride lo
    g1[7] = 0;
    i32x4 gz4 = {0, 0, 0, 0};
    i32x8 gz8 = {0, 0, 0, 0, 0, 0, 0, 0};
    // 6-arg toolchain variant: (g0, g1, g2, g3, extra_x8, cpol)
    __builtin_amdgcn_tensor_load_to_lds(g0, g1, gz4, gz4, gz8, 0);
#if __has_builtin(__builtin_amdgcn_s_wait_tensorcnt)
    __builtin_amdgcn_s_wait_tensorcnt(0);
#endif
  }
#else
  for (int t = tid; t < L_N; t += TPB)
    r0[t] = seasonal[((size_t)b * L_N + t) * C_N + c];
#endif

  // --- forward FFT ---------------------------------------------------------
  fft8192(r0, i0, r1, i1, fAr, fAiN, fAi, lane, wave, tid);

  // --- Hilbert multiplier fused with conjugation (ifft via forward fft) ---
  for (int t = tid; t < L_N; t += TPB) {
    const float h = (t == 0 || t == L_N / 2) ? 1.f : ((t < L_N / 2) ? 2.f : 0.f);
    r0[t] = h * r0[t];
    i0[t] = -h * i0[t];
  }
  fft8192(r0, i0, r1, i1, fAr, fAiN, fAi, lane, wave, tid);

  // --- phase = angle(conj(F)/N) = atan2(-Im, Re) ---------------------------
  for (int t = tid; t < L_N; t += TPB) r1[t] = atan2f(-i0[t], r0[t]);
  __syncthreads();

  // --- phase unwrap: workgroup prefix scan over wrapped diffs --------------
  {
    const float* P = r1;
    float* U = i1;
    const int chunk = L_N / TPB;  // 32
    const int t0 = tid * chunk;
    float partial = 0.f;
    for (int ss = t0; ss < t0 + chunk; ++ss)
      if (ss > 0) partial += wrap_dm(P[ss] - P[ss - 1]);
    scanb[tid] = partial;
    __syncthreads();
    for (int off = 1; off < TPB; off <<= 1) {  // Hillis-Steele inclusive scan
      const float add = (tid >= off) ? scanb[tid - off] : 0.f;
      __syncthreads();
      scanb[tid] += add;
      __syncthreads();
    }
    const float excl = (tid == 0) ? 0.f : scanb[tid - 1];
    const float p00 = P[0];
    float run = p00 + excl;
    for (int ss = t0; ss < t0 + chunk; ++ss) {
      if (ss == 0) {
        U[0] = p00;
      } else {
        run += wrap_dm(P[ss] - P[ss - 1]);
        U[ss] = run;
      }
    }
  }
  __syncthreads();

  // --- conv (reflect pad) + phi0 + AM envelope + cos, coalesced store -----
  const float eps = 1e-6f;
  const float tanhS = tanhf(strength[0]);
  const float beta1 = softplusf(b1l[0]) + eps;
  const float beta2 = softplusf(b2l[0]) + eps;
  const float alpha = softplusf(alpha_log[c]) + eps;
  const float p0c = phi0[c];
  const float* U = i1;
  float* outp = out + (size_t)blockIdx.x * L_N;  // out[b][c][:]
  for (int t = tid; t < L_N; t += TPB) {
    float acc = 0.f;
#pragma unroll
    for (int jj = 0; jj < K_N; ++jj) {
      int m = t + jj - (K_N / 2);
      m = (m < 0) ? -m : m;
      m = (m >= L_N) ? (2 * L_N - 2 - m) : m;
      acc += wz[jj] * U[m];
    }
    const float phi = U[t] + tanhS * acc + p0c;
    float T = trend[(size_t)t * C_N + c];  // batch 0 only (faithful)
    T = fminf(10.f, fmaxf(-10.f, T));
    const float A = alpha * beta1 * log1pf(expf(beta2 * T));
    outp[t] = A * cosf(phi);
  }
}

extern "C" void kernel_launch(void* const* d_in, const int* in_sizes, int n_in,
                              void* d_out, int out_size, void* d_ws,
                              size_t ws_size, hipStream_t stream) {
  (void)in_sizes; (void)n_in; (void)out_size; (void)d_ws; (void)ws_size;
  const float* seasonal = (const float*)d_in[0];
  const float* trend = (const float*)d_in[1];
  const float* pcw = (const float*)d_in[2];
  const float* strength = (const float*)d_in[3];
  const float* alpha_lg = (const float*)d_in[4];
  const float* phi0 = (const float*)d_in[5];
  const float* b1l = (const float*)d_in[6];
  const float* b2l = (const float*)d_in[7];
  float* out = (float*)d_out;

  const size_t shmem = (size_t)33040 * sizeof(float);  // ~129 KB dynamic LDS
  (void)hipFuncSetAttribute((const void*)tsc_kernel,
                            hipFuncAttributeMaxDynamicSharedMemorySize,
                            (int)shmem);
  tsc_kernel<<<dim3(B_N * C_N), dim3(TPB), shmem, stream>>>(
      seasonal, trend, pcw, strength, alpha_lg, phi0, b1l, b2l, out);
}